// XPBDStepBatch_90237262889314
// MI455X (gfx1250) — compile-verified
//
#include <hip/hip_runtime.h>
#include <hip/hip_bf16.h>
#include <math.h>

#define B_      8
#define ITER_   10
#define DT_     0.01f
#define PLANE_H_ (-2.0f)
#define EPS_    1e-5f

typedef unsigned int u32x4 __attribute__((ext_vector_type(4)));
typedef int          i32x4 __attribute__((ext_vector_type(4)));
typedef int          i32x8 __attribute__((ext_vector_type(8)));

// Non-returning device-scope f32 atomic add: guarantees the single
// global_atomic_add_f32 HW instruction (L2 atomic unit), never a CAS loop.
__device__ __forceinline__ void atom_add_f32(float* p, float v) {
    asm volatile("global_atomic_add_f32 %0, %1, off scope:SCOPE_DEV"
                 :: "v"(p), "v"(v) : "memory");
}

// ---------------- predict: Vp = V + dt*(vel + dt*F/m) ----------------
__global__ __launch_bounds__(256)
void k_predict(const float* __restrict__ V, const float* __restrict__ vel,
               const float* __restrict__ F, const float* __restrict__ mass,
               float* __restrict__ Vp, int total /*B*N*/) {
    int p = blockIdx.x * blockDim.x + threadIdx.x;
    if (p >= total) return;
    float s = DT_ / mass[p];
    size_t o = (size_t)p * 3;
    Vp[o + 0] = V[o + 0] + DT_ * (vel[o + 0] + s * F[o + 0]);
    Vp[o + 1] = V[o + 1] + DT_ * (vel[o + 1] + s * F[o + 1]);
    Vp[o + 2] = V[o + 2] + DT_ * (vel[o + 2] + s * F[o + 2]);
}

// ---- precompute per-(b,e) constants: {w_i, w_j, A, S_eff} (time-invariant) ----
__global__ __launch_bounds__(256)
void k_pre(const float* __restrict__ Vw, const float* __restrict__ Vcomp,
           const int2* __restrict__ cdist, float4* __restrict__ ws4,
           int N, int E) {
    int e = blockIdx.x * blockDim.x + threadIdx.x;
    if (e >= E) return;
    int2 ij = cdist[e];
#pragma unroll
    for (int b = 0; b < B_; ++b) {
        size_t bi = (size_t)b * N + ij.x;
        size_t bj = (size_t)b * N + ij.y;
        float wi = Vw[bi], wj = Vw[bj];
        float A  = 0.5f * (Vcomp[bi] + Vcomp[bj]);
        float S  = wi + wj;
        if (S == 0.0f) S = INFINITY;
        float4 q; q.x = wi; q.y = wj; q.z = A; q.w = S;
        ws4[(size_t)b * E + e] = q;
    }
}

// ---------------- Jacobi distance-constraint projection ----------------
__global__ __launch_bounds__(256)
void k_constraint(const float* __restrict__ Vp, float* __restrict__ delta,
                  float* __restrict__ Lbuf, const float4* __restrict__ ws4,
                  const float* __restrict__ Vw, const float* __restrict__ Vcomp,
                  const int2* __restrict__ cdist, const float* __restrict__ initd,
                  int N, int E, int use_pre) {
    int e = blockIdx.x * blockDim.x + threadIdx.x;
    if (e >= E) return;
    int2  ij = cdist[e];
    float d0 = initd[e];
#pragma unroll
    for (int b = 0; b < B_; ++b) {
        size_t bi = (size_t)b * N + ij.x;
        size_t bj = (size_t)b * N + ij.y;
        const float* pi = Vp + bi * 3;
        const float* pj = Vp + bj * 3;
        float nx = pi[0] - pj[0];
        float ny = pi[1] - pj[1];
        float nz = pi[2] - pj[2];
        float D   = sqrtf(nx * nx + ny * ny + nz * nz);
        float inv = 1.0f / D;          // matches reference (inf/nan if D==0)
        float C   = D - d0;

        float wi, wj, A, S;
        if (use_pre) {
            float4 q = ws4[(size_t)b * E + e];
            wi = q.x; wj = q.y; A = q.z; S = q.w;
        } else {
            wi = Vw[bi]; wj = Vw[bj];
            A  = 0.5f * (Vcomp[bi] + Vcomp[bj]);
            S  = wi + wj;
            if (S == 0.0f) S = INFINITY;
        }

        size_t le = (size_t)b * E + e;
        float L  = Lbuf[le];
        float Ld = (-C - A * L) / (S + A);
        Lbuf[le] = L + Ld;

        float s  = Ld * inv;
        float ux = s * nx, uy = s * ny, uz = s * nz;
        float* di = delta + bi * 3;
        float* dj = delta + bj * 3;
        atom_add_f32(di + 0,  wi * ux);
        atom_add_f32(di + 1,  wi * uy);
        atom_add_f32(di + 2,  wi * uz);
        atom_add_f32(dj + 0, -wj * ux);
        atom_add_f32(dj + 1, -wj * uy);
        atom_add_f32(dj + 2, -wj * uz);
    }
}

// ---- apply: Vp += delta; delta = 0.
// Vp tile staged per-lane via global_load_async_to_lds_b128 (ASYNCcnt);
// delta tile moved by a single Tensor Data Mover DMA per block (TENSORcnt). ----
__global__ __launch_bounds__(256)
void k_apply(float* __restrict__ Vp, float* __restrict__ delta, int total4) {
    __shared__ __align__(16) float s_vp[256 * 4];
    __shared__ __align__(16) float s_delta[256 * 4];
    int idx = blockIdx.x * blockDim.x + threadIdx.x;
    bool valid = idx < total4;

    if (valid) {
        unsigned lds = (unsigned)(unsigned long long)&s_vp[threadIdx.x * 4];
        const float* g = Vp + (size_t)idx * 4;
        asm volatile("global_load_async_to_lds_b128 %0, %1, off"
                     :: "v"(lds), "v"(g) : "memory");
    }

    if (threadIdx.x < 32) {  // wave 0 issues one TDM op for the 4KB delta tile
        unsigned lds_d = (unsigned)(unsigned long long)&s_delta[0];
        unsigned long long gaddr =
            (unsigned long long)(const void*)delta + (unsigned long long)blockIdx.x * 4096ull;
        // D# group 0: count=1, lds_addr, global_addr, type=2 (image)
        u32x4 g0;
        g0[0] = 1u;
        g0[1] = lds_d;
        g0[2] = (unsigned)gaddr;
        g0[3] = ((unsigned)(gaddr >> 32) & 0x01FFFFFFu) | (2u << 30);
        // D# group 1: data_size=3 (8B), tensor_dim0=512, tensor_dim1=1,
        //             tile_dim0=512, tile_dim1=1, tensor_dim0_stride=512
        i32x8 g1;
        g1[0] = 0x00030000; g1[1] = 0x02000000; g1[2] = 0x00010000; g1[3] = 0x02000000;
        g1[4] = 1;          g1[5] = 512;        g1[6] = 0;          g1[7] = 0;
        // D# group 2: tensor_dim2=1, tensor_dim3=1, stride2=0, tile_dim3=0
        i32x4 g2; g2[0] = 1; g2[1] = 1; g2[2] = 0; g2[3] = 0;
        // D# group 3: stride3=0, tensor_dim4=1, tile_dim4=0
        i32x4 g3; g3[0] = 0; g3[1] = 0x00010000; g3[2] = 0; g3[3] = 0;
#if defined(__clang_major__) && (__clang_major__ >= 23)
        i32x8 g4 = {0, 0, 0, 0, 0, 0, 0, 0};
        __builtin_amdgcn_tensor_load_to_lds(g0, g1, g2, g3, g4, 0);
#else
        __builtin_amdgcn_tensor_load_to_lds(g0, g1, g2, g3, 0);
#endif
        __builtin_amdgcn_s_wait_tensorcnt(0);
    }

    asm volatile("s_wait_asynccnt 0x0" ::: "memory");
    __syncthreads();

    if (valid) {
        float4 d = *(const float4*)&s_delta[threadIdx.x * 4];
        float4 v = *(const float4*)&s_vp[threadIdx.x * 4];
        float4 r;
        r.x = v.x + d.x; r.y = v.y + d.y; r.z = v.z + d.z; r.w = v.w + d.w;
        ((float4*)Vp)[idx] = r;
        float4 z; z.x = 0.f; z.y = 0.f; z.z = 0.f; z.w = 0.f;
        ((float4*)delta)[idx] = z;
    }
}

// ---------------- finalize: fold last delta + plane collision + velocity ----------------
__global__ __launch_bounds__(256)
void k_finalize(const float* __restrict__ V0, const float* __restrict__ delta,
                float* __restrict__ out, int total /*B*N*/, int BN3) {
    int p = blockIdx.x * blockDim.x + threadIdx.x;
    if (p >= total) return;
    size_t o = (size_t)p * 3;
    float vx = V0[o + 0], vy = V0[o + 1], vz0 = V0[o + 2];
    float px = out[o + 0] + delta[o + 0];
    float py = out[o + 1] + delta[o + 1];
    float pz = out[o + 2] + delta[o + 2];

    bool  col     = (pz < PLANE_H_) && (vz0 > PLANE_H_);
    float h_prev  = vz0 - PLANE_H_;
    float h_after = PLANE_H_ - pz;
    float denom   = col ? (h_prev + h_after) : 1.0f;
    float t       = h_prev / denom;
    float xx = col ? (vx + t * (px - vx)) : px;
    float yy = col ? (vy + t * (py - vy)) : py;
    float z1 = col ? (PLANE_H_ + EPS_) : pz;
    bool  vio = (z1 < PLANE_H_) && (vz0 < PLANE_H_);
    float z2  = vio ? (PLANE_H_ + EPS_) : z1;

    float invdt = 1.0f / DT_;
    float ax = (xx - vx) * invdt;
    float ay = (yy - vy) * invdt;
    float az = (z2 - vz0) * invdt;
    az = col ? -az : az;
    az = vio ? 0.0f : az;

    out[o + 0] = xx; out[o + 1] = yy; out[o + 2] = z2;
    out[BN3 + o + 0] = ax; out[BN3 + o + 1] = ay; out[BN3 + o + 2] = az;
}

extern "C" void kernel_launch(void* const* d_in, const int* in_sizes, int n_in,
                              void* d_out, int out_size, void* d_ws, size_t ws_size,
                              hipStream_t stream) {
    const float* V     = (const float*)d_in[0];
    const float* Vvel  = (const float*)d_in[1];
    const float* Vw    = (const float*)d_in[2];
    const float* Vcomp = (const float*)d_in[3];
    const float* Vmass = (const float*)d_in[4];
    const float* Vforce= (const float*)d_in[5];
    const float* Cinit = (const float*)d_in[6];
    const int2*  Cdist = (const int2*)d_in[7];

    const int BN3 = in_sizes[0];          // B*N*3
    const int BN  = in_sizes[4];          // B*N
    const int N   = BN / B_;
    const int E   = in_sizes[6];          // C_init_d element count

    float* out   = (float*)d_out;         // [Vp / final V | final velocity]
    float* delta = (float*)d_ws;          // BN3 floats
    float* Lbuf  = delta + BN3;           // B*E floats
    float4* ws4  = (float4*)(Lbuf + (size_t)B_ * E);

    size_t base_bytes = ((size_t)BN3 + (size_t)B_ * E) * sizeof(float);
    size_t pre_bytes  = (size_t)B_ * E * sizeof(float4);
    int use_pre = (ws_size >= base_bytes + pre_bytes) ? 1 : 0;

    hipMemsetAsync(d_ws, 0, base_bytes, stream);   // delta = 0, L = 0

    const int T = 256;
    k_predict<<<(BN + T - 1) / T, T, 0, stream>>>(V, Vvel, Vforce, Vmass, out, BN);
    if (use_pre)
        k_pre<<<(E + T - 1) / T, T, 0, stream>>>(Vw, Vcomp, Cdist, ws4, N, E);

    const int total4 = BN3 >> 2;          // BN3 = 24*N, divisible by 4
    for (int it = 0; it < ITER_; ++it) {
        k_constraint<<<(E + T - 1) / T, T, 0, stream>>>(
            out, delta, Lbuf, ws4, Vw, Vcomp, Cdist, Cinit, N, E, use_pre);
        if (it != ITER_ - 1)
            k_apply<<<(total4 + T - 1) / T, T, 0, stream>>>(out, delta, total4);
    }

    k_finalize<<<(BN + T - 1) / T, T, 0, stream>>>(V, delta, out, BN, BN3);
}